// ProbasLinear_42116449304820
// MI455X (gfx1250) — compile-verified
//
#include <hip/hip_runtime.h>

#define B_    1024
#define OUT_  512
#define IN_   512
#define TB    8     // batch rows per block (one per wave)
#define TO    8     // output cols per block (sigma rows staged in LDS)

typedef __attribute__((ext_vector_type(2))) float v2f;
typedef __attribute__((ext_vector_type(4))) float v4f;
typedef __attribute__((ext_vector_type(8))) float v8f;

// ---------------------------------------------------------------------------
// Kernel A: streams eps_w (1 GiB, the entire cost of this op) with NT hints.
// Block = 8 waves covers an (8 b x 8 o) tile:
//   - |sigma| rows (8 x 2 KiB) staged once into LDS (fabs folded in)
//   - each wave pins its x row (16 floats/lane) in VGPRs
//   - inner loop: eps NT global_load_b128 + sigma ds_load_b128 + FMA chain
// L2 refetch traffic drops from ~2 GiB (naive) to ~256 MB; eps stays the
// only HBM stream and is marked non-temporal so it can't evict x/mu/sigma.
// ---------------------------------------------------------------------------
__global__ void probas_eps_stream(const float* __restrict__ x,
                                  const float* __restrict__ wsig,
                                  const float* __restrict__ bmu,
                                  const float* __restrict__ bsig,
                                  const float* __restrict__ eps_w,
                                  const float* __restrict__ eps_b,
                                  float* __restrict__ out) {
    __shared__ float s_abs[TO * IN_];          // 16 KiB of |sigma| rows

    const int tid  = threadIdx.x;
    const int lane = tid & 31;
    const int wv   = tid >> 5;                 // wave index in block: 0..7
    const int blocksPerB = OUT_ / TO;          // 64
    const int b0 = (blockIdx.x / blocksPerB) * TB;
    const int o0 = (blockIdx.x % blocksPerB) * TO;

    // --- Stage |sigma| tile into LDS (coalesced, 4 v4f per thread) ---------
#pragma unroll
    for (int idx = tid * 4; idx < TO * IN_; idx += 256 * 4) {
        const int oo = idx >> 9;               // idx / IN_
        const int ii = idx & (IN_ - 1);        // idx % IN_
        v4f s = *(const v4f*)(wsig + (size_t)(o0 + oo) * IN_ + ii);
        v4f a;
        a.x = __builtin_fabsf(s.x);
        a.y = __builtin_fabsf(s.y);
        a.z = __builtin_fabsf(s.z);
        a.w = __builtin_fabsf(s.w);
        *(v4f*)(s_abs + idx) = a;
    }
    __syncthreads();

    // --- Pin this wave's x row in registers (16 floats/lane) ---------------
    const int b = b0 + wv;
    v4f xr[4];
#pragma unroll
    for (int it = 0; it < 4; ++it)
        xr[it] = *(const v4f*)(x + (size_t)b * IN_ + it * 128 + lane * 4);

    const float* __restrict__ erow0 = eps_w + ((size_t)b * OUT_ + o0) * IN_;

    // --- 8 eps rows: stream, scale, reduce ---------------------------------
#pragma unroll 2
    for (int j = 0; j < TO; ++j) {
        const float* __restrict__ er = erow0 + (size_t)j * IN_;
        const float* __restrict__ sr = s_abs + j * IN_;
        float acc = 0.0f;
#pragma unroll
        for (int it = 0; it < 4; ++it) {
            const int i = it * 128 + lane * 4;
            v4f ee = __builtin_nontemporal_load((const v4f*)(er + i)); // NT HBM stream
            v4f ss = *(const v4f*)(sr + i);                            // ds_load_b128
            acc = __builtin_fmaf(xr[it].x * ss.x, ee.x, acc);
            acc = __builtin_fmaf(xr[it].y * ss.y, ee.y, acc);
            acc = __builtin_fmaf(xr[it].z * ss.z, ee.z, acc);
            acc = __builtin_fmaf(xr[it].w * ss.w, ee.w, acc);
        }
        // wave32 butterfly reduction
#pragma unroll
        for (int off = 16; off > 0; off >>= 1)
            acc += __shfl_xor(acc, off, 32);

        if (lane == 0) {
            const int o = o0 + j;
            const size_t idx = (size_t)b * OUT_ + o;
            const float eb = __builtin_nontemporal_load(eps_b + idx);
            out[idx] = acc + bmu[o] + __builtin_fabsf(bsig[o]) * eb;
        }
    }
}

// ---------------------------------------------------------------------------
// Kernel B: x[B,IN] @ mu[OUT,IN]^T added into out via V_WMMA_F32_16X16X4_F32.
// One wave per 16x16 output tile; K loop of 128 WMMAs at full fp32 precision.
// C is seeded from d_out (eps+bias partial), so the final store is the answer.
//
// Layouts (ISA 7.12.2, 32-bit operands, wave32):
//   A 16x4:  lane -> M = lane&15 ; VGPR0/1 -> K = (lane>>4)*2 + {0,1}
//   B 4x16:  lane -> N = lane&15 ; VGPR0/1 -> K = (lane>>4)*2 + {0,1}
//   C 16x16: VGPR r, lanes 0-15: M=r, N=lane ; lanes 16-31: M=8+r, N=lane-16
// ---------------------------------------------------------------------------
__global__ void probas_wmma_gemm(const float* __restrict__ x,
                                 const float* __restrict__ wmu,
                                 float* __restrict__ out) {
    const int lane   = threadIdx.x & 31;
    const int waveId = (int)((blockIdx.x * blockDim.x + threadIdx.x) >> 5);
    const int tilesN = OUT_ / 16;                  // 32
    const int b0 = (waveId / tilesN) * 16;
    const int o0 = (waveId % tilesN) * 16;

    const int half = lane >> 4;                    // 0 or 1
    const int l15  = lane & 15;
    const int koff = half * 2;

    // Load C accumulator tile from the eps/bias partial already in out.
    const int crow0 = b0 + half * 8;
    const int ccol  = o0 + l15;
    v8f c;
#pragma unroll
    for (int r = 0; r < 8; ++r)
        c[r] = out[(size_t)(crow0 + r) * OUT_ + ccol];

    const float* __restrict__ arow = x   + (size_t)(b0 + l15) * IN_; // A: x row M
    const float* __restrict__ brow = wmu + (size_t)(o0 + l15) * IN_; // B: mu row N (B[k][n]=mu[n][k])

#pragma unroll 4
    for (int k0 = 0; k0 < IN_; k0 += 4) {
        v2f a = *(const v2f*)(arow + k0 + koff);   // 8B aligned (koff even)
        v2f b = *(const v2f*)(brow + k0 + koff);
        // 8 args: (neg_a, A, neg_b, B, c_mod, C, reuse_a, reuse_b)
        c = __builtin_amdgcn_wmma_f32_16x16x4_f32(false, a, false, b,
                                                  (short)0, c, false, false);
    }

#pragma unroll
    for (int r = 0; r < 8; ++r)
        out[(size_t)(crow0 + r) * OUT_ + ccol] = c[r];
}

// ---------------------------------------------------------------------------
extern "C" void kernel_launch(void* const* d_in, const int* in_sizes, int n_in,
                              void* d_out, int out_size, void* d_ws, size_t ws_size,
                              hipStream_t stream) {
    const float* x     = (const float*)d_in[0];   // [B, IN]
    const float* wmu   = (const float*)d_in[1];   // [OUT, IN]
    const float* wsig  = (const float*)d_in[2];   // [OUT, IN]
    const float* bmu   = (const float*)d_in[3];   // [OUT]
    const float* bsig  = (const float*)d_in[4];   // [OUT]
    const float* eps_w = (const float*)d_in[5];   // [B, OUT, IN]
    const float* eps_b = (const float*)d_in[6];   // [B, OUT]
    float* out = (float*)d_out;                   // [B, OUT]

    // Kernel A: (B/8)*(OUT/8) = 8192 blocks of 8 waves; writes out = eps+bias.
    {
        const int blocks = (B_ / TB) * (OUT_ / TO);     // 8192
        probas_eps_stream<<<blocks, 256, 0, stream>>>(x, wsig, bmu, bsig,
                                                      eps_w, eps_b, out);
    }
    // Kernel B: one wave per 16x16 tile -> 2048 waves; out += x @ mu^T.
    {
        const int totalWaves = (B_ / 16) * (OUT_ / 16); // 2048
        const int blocks = totalWaves / 8;              // 256
        probas_wmma_gemm<<<blocks, 256, 0, stream>>>(x, wmu, out);
    }
}